// Outer1DTo2DLayer_16870631538966
// MI455X (gfx1250) — compile-verified
//
#include <hip/hip_runtime.h>
#include <stdint.h>

// Problem constants from the reference (F and PAD are fixed by the model).
#define NFEAT   50      // N_FEATURES
#define PADLEN  25      // PAD_LENGTH
#define QTILE   128     // q-rows per block tile

typedef float vf4 __attribute__((ext_vector_type(4)));  // native vector: OK for NT builtin

// out[p, q, 0:50]   = feats[p, :]          if flag[p,q] else 0
// out[p, q, 50:100] = feats[base[p]+q, :]  if flag[p,q] else 0
//
// Store-bound kernel (324 MB out vs ~3.5 MB in). CDNA5 path used:
//   - global_load_async_to_lds_b64 (ASYNCcnt) streams the gather slab to LDS
//   - s_wait_asynccnt 0 + workgroup barrier orders LDS visibility
//   - b128 non-temporal stores (write-once stream larger than the 192MB L2)
__global__ __launch_bounds__(256)
void outer1d_to_2d_kernel(const float* __restrict__ feats,  // [total, NFEAT]
                          const int*   __restrict__ flag,   // [total, maxq]
                          const int*   __restrict__ nraw,   // [nsamp]
                          float*       __restrict__ out,    // [total, maxq, 2*NFEAT]
                          int total, int maxq, int nsamp)
{
    __shared__ float sP[NFEAT];          // feats[p, :]
    __shared__ int   sFlag[QTILE];       // flag[p, q0:q0+rows]
    __shared__ int   sBase;              // base[p]
    __shared__ float s2[QTILE * NFEAT];  // feats[base+q0 : base+q0+rows, :]

    const int p   = blockIdx.x;
    const int q0  = blockIdx.y * QTILE;
    const int tid = threadIdx.x;

    // --- base[p]: prefix-sum walk over (nraw[s] + PADLEN); tiny scalar loop ---
    if (tid == 0) {
        int off = 0, b = 0;
        for (int s = 0; s < nsamp; ++s) {
            int len = nraw[s] + PADLEN;
            if (p < off + len) { b = off; break; }
            off += len;
        }
        sBase = b;
    }

    // --- stage broadcast row and flag row (tiny; plain loads, stay cached) ---
    if (tid < NFEAT) sP[tid] = feats[(size_t)p * NFEAT + tid];
    int rows = maxq - q0;
    if (rows > QTILE) rows = QTILE;
    if (tid < rows) sFlag[tid] = flag[(size_t)p * maxq + q0 + tid];
    __syncthreads();                       // sBase/sP/sFlag visible
    const int base = sBase;

    // --- async-stage the gather slab: consecutive q map to consecutive feats
    //     rows, so the [rows x NFEAT] tile is one flat region. 8-byte granules
    //     (row stride 200B guarantees only 8B alignment). Clamp to the packed
    //     array end; clamped-off rows are exactly the flag==0 rows.
    int srcRow   = base + q0;
    int copyRows = total - srcRow;
    if (copyRows < 0)    copyRows = 0;
    if (copyRows > rows) copyRows = rows;
    const int nF2 = copyRows * (NFEAT / 2);                 // float2 count
    const uint64_t gsrc = (uint64_t)(uintptr_t)(feats + (size_t)srcRow * NFEAT);
    for (int c = tid; c < nF2; c += 256) {
        uint32_t ldsoff = (uint32_t)(uintptr_t)(&s2[c * 2]); // AS(3) ptr -> LDS byte offset
        uint32_t voff   = (uint32_t)(c * 8);
        asm volatile("global_load_async_to_lds_b64 %0, %1, %2"
                     :: "v"(ldsoff), "v"(voff), "s"(gsrc)
                     : "memory");
    }
    asm volatile("s_wait_asynccnt 0" ::: "memory");          // my wave's copies done
    __syncthreads();                                         // everyone's copies visible

    // --- compose + non-temporal b128 stores: each output row = 25 vf4 chunks ---
    const int nChunks = rows * 25;                           // (2*NFEAT)/4 == 25
    vf4* __restrict__ out4 = (vf4*)out;
    for (int c = tid; c < nChunks; c += 256) {
        const int row = c / 25;
        const int rem = c - row * 25;
        const int fo  = rem * 4;                             // first f of this chunk
        const int fl  = sFlag[row];
        vf4 v;
        if (fo < NFEAT - 2) {            // fo <= 44: pure broadcast half
            v.x = sP[fo];     v.y = sP[fo + 1];
            v.z = sP[fo + 2]; v.w = sP[fo + 3];
        } else if (fo == NFEAT - 2) {    // fo == 48: straddles the concat seam
            v.x = sP[NFEAT - 2];         v.y = sP[NFEAT - 1];
            v.z = s2[row * NFEAT + 0];   v.w = s2[row * NFEAT + 1];
        } else {                          // fo >= 52: pure gather half
            const int k = fo - NFEAT;    // 2,6,...,46
            v.x = s2[row * NFEAT + k];     v.y = s2[row * NFEAT + k + 1];
            v.z = s2[row * NFEAT + k + 2]; v.w = s2[row * NFEAT + k + 3];
        }
        // Select (not multiply): uninitialized LDS may hold NaN; 0*NaN != 0.
        if (!fl) { v = (vf4)0.0f; }
        // Streaming write-once output: NT hint keeps it out of L2's way.
        __builtin_nontemporal_store(v, &out4[((size_t)p * maxq + (q0 + row)) * 25 + rem]);
    }
}

extern "C" void kernel_launch(void* const* d_in, const int* in_sizes, int n_in,
                              void* d_out, int out_size, void* d_ws, size_t ws_size,
                              hipStream_t stream) {
    const float* feats = (const float*)d_in[0];   // [1, total, 50] f32
    const int*   flag  = (const int*)d_in[1];     // [1, total, maxq] i32
    const int*   nraw  = (const int*)d_in[2];     // [nsamp] i32
    float*       out   = (float*)d_out;           // [1, total, maxq, 100] f32

    const int nsamp = in_sizes[2];
    const int total = in_sizes[0] / NFEAT;        // 1350
    const int maxq  = in_sizes[1] / total;        // 600

    dim3 grid((unsigned)total, (unsigned)((maxq + QTILE - 1) / QTILE));
    outer1d_to_2d_kernel<<<grid, 256, 0, stream>>>(feats, flag, nraw, out,
                                                   total, maxq, nsamp);
}